// Kuramoto_global_18425409699990
// MI455X (gfx1250) — compile-verified
//
#include <hip/hip_runtime.h>
#include <math.h>

#define N_NODES 50000
#define N_EDGES 1250000
#define D_FEAT  64

typedef __attribute__((ext_vector_type(2))) float v2f;
typedef __attribute__((ext_vector_type(8))) float v8f;

__device__ __forceinline__ float wave_sum(float v) {
  v += __shfl_xor(v, 16, 32);
  v += __shfl_xor(v, 8, 32);
  v += __shfl_xor(v, 4, 32);
  v += __shfl_xor(v, 2, 32);
  v += __shfl_xor(v, 1, 32);
  return v;
}

__device__ __forceinline__ void atomic_add_f32(float* p, float v) {
  __hip_atomic_fetch_add(p, v, __ATOMIC_RELAXED, __HIP_MEMORY_SCOPE_AGENT);
}

// Kernel 1: one wave32 per node row. Normalize state row, zero accumulator.
__global__ __launch_bounds__(256) void k_norm_zero(const float* __restrict__ state,
                                                   float* __restrict__ norm,
                                                   float* __restrict__ acc) {
  int node = blockIdx.x * 8 + (threadIdx.x >> 5);
  int L = threadIdx.x & 31;
  if (node >= N_NODES) return;
  int base = node * D_FEAT + 2 * L;
  v2f s = *(const v2f*)(state + base);
  float ss = wave_sum(s.x * s.x + s.y * s.y);
  float inv = 1.0f / sqrtf(ss);
  v2f n; n.x = s.x * inv; n.y = s.y * inv;
  *(v2f*)(norm + base) = n;
  v2f z; z.x = 0.0f; z.y = 0.0f;
  *(v2f*)(acc + base) = z;
}

// Kernel 2: one wave32 per tile of 16 edges.
//   - 16 edge dots = diagonal of 16x16 f32 GEMM via V_WMMA_F32_16X16X4_F32 (K=64 in 16 steps)
//   - per-edge gelu MLP split across lane pairs (L, L+16)
//   - scatter 128 f32 atomics per edge into L2-resident accumulator
__global__ __launch_bounds__(256) void k_edges(const float* __restrict__ norm,
                                               const int*   __restrict__ ind,
                                               const float* __restrict__ w1,
                                               const float* __restrict__ b1,
                                               const float* __restrict__ w2,
                                               const float* __restrict__ b2,
                                               float* __restrict__ acc) {
  __shared__ float lw1[64], lb1[64], lw2[64];
  int t = threadIdx.x;
  if (t < 64) { lw1[t] = w1[t]; lb1[t] = b1[t]; lw2[t] = w2[t]; }
  __syncthreads();

  const int tiles = N_EDGES / 16;            // 78125, exact
  int wave = blockIdx.x * 8 + (t >> 5);
  if (wave >= tiles) return;                 // wave-uniform: EXEC stays all-ones for WMMA

  int L = t & 31;
  int m = L & 15;                            // edge-in-tile this lane serves in the fragments
  int e = wave * 16 + m;
  int sBase = ind[2 * e]     * D_FEAT;       // src row offset (floats)
  int dBase = ind[2 * e + 1] * D_FEAT;       // dst row offset
  int koff  = (L >> 4) << 1;                 // lanes 0-15: K+0/+1, lanes 16-31: K+2/+3

  const float* sp = norm + sBase + koff;     // A fragment source (16x4 per step)
  const float* dp = norm + dBase + koff;     // B fragment source (4x16 per step, = dst rows)

  v8f cacc = {};
#pragma unroll
  for (int k = 0; k < 16; ++k) {
    v2f a = *(const v2f*)(sp + 4 * k);
    v2f bfrag = *(const v2f*)(dp + 4 * k);
    cacc = __builtin_amdgcn_wmma_f32_16x16x4_f32(false, a, false, bfrag,
                                                 (short)0, cacc, false, false);
  }

  // Diagonal C[e][e]: VGPR (e&7), lane (e<8 ? e : e+8). Select then shuffle to lanes m, m+16.
  int r = L & 7;
  float dval = cacc[0];
  dval = (r == 1) ? cacc[1] : dval;
  dval = (r == 2) ? cacc[2] : dval;
  dval = (r == 3) ? cacc[3] : dval;
  dval = (r == 4) ? cacc[4] : dval;
  dval = (r == 5) ? cacc[5] : dval;
  dval = (r == 6) ? cacc[6] : dval;
  dval = (r == 7) ? cacc[7] : dval;
  int srcLane = (m < 8) ? m : (m + 8);
  float s = __shfl(dval, srcLane, 32);       // edge m's dot, on lanes m and m+16

  // Hidden layer: lane L handles 32 of the 64 hidden units for edge m.
  int jb = (L >> 4) * 32;
  float hsum = 0.0f;
#pragma unroll 4
  for (int j = 0; j < 32; ++j) {
    float w1j = lw1[jb + j], b1j = lb1[jb + j], w2j = lw2[jb + j];
    float x = fmaf(s, w1j, b1j);
    float inner = 0.79788456080286536f * fmaf(0.044715f * x, x * x, x);
    float h = 0.5f * x * (1.0f + tanhf(inner));     // jax.nn.gelu (tanh approx)
    hsum = fmaf(h, w2j, hsum);
  }
  float dE = hsum + __shfl_xor(hsum, 16, 32) + b2[0];  // valid on lanes m and m+16

  // Scatter: whole wave cooperates per edge; lane L covers features 2L, 2L+1.
  int f = 2 * L;
#pragma unroll 4
  for (int eo = 0; eo < 16; ++eo) {
    float de = __shfl(dE, eo, 32);
    int sO = __shfl(sBase, eo, 32);
    int dO = __shfl(dBase, eo, 32);
    v2f sv = *(const v2f*)(norm + sO + f);
    v2f dv = *(const v2f*)(norm + dO + f);
    atomic_add_f32(acc + sO + f,     de * dv.x);
    atomic_add_f32(acc + sO + f + 1, de * dv.y);
    atomic_add_f32(acc + dO + f,     de * sv.x);
    atomic_add_f32(acc + dO + f + 1, de * sv.y);
  }
}

// Kernel 3: out = -acc + n * (n . acc), one wave32 per node.
__global__ __launch_bounds__(256) void k_finalize(const float* __restrict__ norm,
                                                  const float* __restrict__ acc,
                                                  float* __restrict__ out) {
  int node = blockIdx.x * 8 + (threadIdx.x >> 5);
  int L = threadIdx.x & 31;
  if (node >= N_NODES) return;
  int base = node * D_FEAT + 2 * L;
  v2f n = *(const v2f*)(norm + base);
  v2f a = *(const v2f*)(acc + base);
  float d = wave_sum(n.x * a.x + n.y * a.y);
  v2f o; o.x = fmaf(n.x, d, -a.x); o.y = fmaf(n.y, d, -a.y);
  *(v2f*)(out + base) = o;
}

extern "C" void kernel_launch(void* const* d_in, const int* in_sizes, int n_in,
                              void* d_out, int out_size, void* d_ws, size_t ws_size,
                              hipStream_t stream) {
  const float* state = (const float*)d_in[0];
  const int*   ind   = (const int*)  d_in[1];
  const float* w1    = (const float*)d_in[2];
  const float* b1    = (const float*)d_in[3];
  const float* w2    = (const float*)d_in[4];
  const float* b2    = (const float*)d_in[5];
  float* out  = (float*)d_out;
  float* norm = (float*)d_ws;                               // 50000*64 f32
  float* acc  = norm + (size_t)N_NODES * D_FEAT;            // 50000*64 f32

  k_norm_zero<<<N_NODES / 8, 256, 0, stream>>>(state, norm, acc);

  int tiles  = N_EDGES / 16;        // 78125
  int blocks = (tiles + 7) / 8;     // 8 waves per 256-thread block
  k_edges<<<blocks, 256, 0, stream>>>(norm, ind, w1, b1, w2, b2, acc);

  k_finalize<<<N_NODES / 8, 256, 0, stream>>>(norm, acc, out);
}